// AdditiveAttention_16312285790467
// MI455X (gfx1250) — compile-verified
//
#include <hip/hip_runtime.h>
#include <hip/hip_bf16.h>
#include <math.h>

// Problem dims (match reference)
#define B_   4
#define Q_   128
#define K_   1024
#define DQ_  512
#define DK_  512
#define DV_  512
#define H_   256
#define NEG_INF_ (-1e6f)

typedef __attribute__((ext_vector_type(2))) float v2f;
typedef __attribute__((ext_vector_type(8))) float v8f;

// CDNA5 hardware V_TANH_F32 (confirmed: picking this path shrank the scores
// kernel ~5x in static VALU count vs the OCML expansion).
#if __has_builtin(__builtin_amdgcn_tanhf)
#define FAST_TANH(x) __builtin_amdgcn_tanhf(x)
#elif __has_builtin(__builtin_amdgcn_tanh_f32)
#define FAST_TANH(x) __builtin_amdgcn_tanh_f32(x)
#else
#define FAST_TANH(x) tanhf(x)
#endif

// ---------------------------------------------------------------------------
// Batched f32 GEMM: C[M,N] = A[M,KD] * B[KD,N], row-major, N/KD compile-time
// so all B/C strides fold into immediate load/store offsets.
// One wave computes a 16x32 tile (two 16x16 WMMA accumulators sharing the
// A fetch) using V_WMMA_F32_16X16X4_F32.
//
// Per ISA layouts (CDNA5 ISA):
//   A 16x4 : lanes 0-15 -> M=lane,   VGPR0=K+0, VGPR1=K+1
//            lanes 16-31-> M=lane-16,VGPR0=K+2, VGPR1=K+3   (contiguous b64 load)
//   B 4x16 : lanes 0-15 -> N=lane,   VGPR0=K+0, VGPR1=K+1   (64B coalesced/halfwave)
//            lanes 16-31-> N=lane-16,VGPR0=K+2, VGPR1=K+3
//   C/D    : VGPR r: lanes 0-15 -> (M=r, N=lane), lanes 16-31 -> (M=r+8, N=lane-16)
// ---------------------------------------------------------------------------
template <int N, int KD>
__global__ void __launch_bounds__(32)
gemm_wmma_f32(const float* __restrict__ A, const float* __restrict__ Bm,
              float* __restrict__ C, long sAb, long sBb, long sCb)
{
  const int lane = threadIdx.x;          // 0..31
  const int half = lane >> 4;            // 0: K pair {0,1}; 1: K pair {2,3}
  const int l    = lane & 15;
  const int m0   = blockIdx.x * 16;
  const int n0   = blockIdx.y * 32;      // two 16-wide N tiles per wave
  const int b    = blockIdx.z;

  const float* arow  = A  + (long)b * sAb + (long)(m0 + l) * KD + half * 2;
  const float* bcol0 = Bm + (long)b * sBb + (long)(half * 2) * N + (n0 + l);
  const float* bcol1 = bcol0 + 16;
  float*       cptr  = C  + (long)b * sCb + (long)(m0 + half * 8) * N + (n0 + l);

  v8f acc0 = {};
  v8f acc1 = {};
  #pragma unroll 8
  for (int k = 0; k < KD; k += 4) {
    __builtin_prefetch(arow + k + 64, 0, 0);   // global_prefetch_b8 on streaming A
    v2f a, b0, b1;
    a.x  = arow[k + 0];                        // contiguous pair -> global_load_b64
    a.y  = arow[k + 1];
    b0.x = bcol0[k * N];                       // constant-stride -> immediate offsets
    b0.y = bcol0[(k + 1) * N];
    b1.x = bcol1[k * N];
    b1.y = bcol1[(k + 1) * N];
    // (neg_a, A, neg_b, B, c_mod, C, reuse_a, reuse_b)
    acc0 = __builtin_amdgcn_wmma_f32_16x16x4_f32(
        false, a, false, b0, (short)0, acc0, false, false);
    acc1 = __builtin_amdgcn_wmma_f32_16x16x4_f32(
        false, a, false, b1, (short)0, acc1, false, false);
  }

  #pragma unroll
  for (int r = 0; r < 8; ++r) {
    cptr[r * N]      = acc0[r];
    cptr[r * N + 16] = acc1[r];
  }
}

// ---------------------------------------------------------------------------
// Fused additive scores + masked softmax.
// One 256-thread block per (b, q). Each thread owns 4 keys: k = tid + j*256.
//   scores[k] = sum_h wv[h] * tanh(qf[b,q,h] + kf[b,k,h]);  mask to -1e6; softmax.
// Dominant phase: B*Q*K*H = 134M tanh -> hardware V_TANH_F32 trans ops.
// ---------------------------------------------------------------------------
__global__ void __launch_bounds__(256)
scores_softmax_kernel(const float* __restrict__ qf, const float* __restrict__ kf,
                      const float* __restrict__ wv, const int* __restrict__ valid_lens,
                      float* __restrict__ attn)
{
  const int q   = blockIdx.x;
  const int b   = blockIdx.y;
  const int tid = threadIdx.x;

  __shared__ float qrow[H_];
  __shared__ float wvs[H_];
  __shared__ float red[8];     // 8 waves of 32

  qrow[tid] = qf[((long)b * Q_ + q) * H_ + tid];
  wvs[tid]  = wv[tid];
  __syncthreads();

  const int vl = valid_lens[b];

  float s[4];
  #pragma unroll
  for (int j = 0; j < 4; ++j) {
    const int k = tid + j * 256;
    const float4* krow = reinterpret_cast<const float4*>(kf + ((long)b * K_ + k) * H_);
    float acc = 0.f;
    #pragma unroll 4
    for (int h4 = 0; h4 < H_ / 4; ++h4) {
      const float4 kv = krow[h4];
      const int h = h4 * 4;
      acc += wvs[h + 0] * FAST_TANH(qrow[h + 0] + kv.x);
      acc += wvs[h + 1] * FAST_TANH(qrow[h + 1] + kv.y);
      acc += wvs[h + 2] * FAST_TANH(qrow[h + 2] + kv.z);
      acc += wvs[h + 3] * FAST_TANH(qrow[h + 3] + kv.w);
    }
    s[j] = (k < vl) ? acc : NEG_INF_;
  }

  // ---- block max (wave32 shuffle + LDS across 8 waves) ----
  float m = fmaxf(fmaxf(s[0], s[1]), fmaxf(s[2], s[3]));
  #pragma unroll
  for (int off = 16; off > 0; off >>= 1)
    m = fmaxf(m, __shfl_xor(m, off, 32));
  if ((tid & 31) == 0) red[tid >> 5] = m;
  __syncthreads();
  float bm = red[0];
  #pragma unroll
  for (int w = 1; w < 8; ++w) bm = fmaxf(bm, red[w]);
  __syncthreads();   // everyone done reading red before reuse

  // ---- block sum of exp ----
  float p[4];
  float lsum = 0.f;
  #pragma unroll
  for (int j = 0; j < 4; ++j) { p[j] = __expf(s[j] - bm); lsum += p[j]; }
  #pragma unroll
  for (int off = 16; off > 0; off >>= 1)
    lsum += __shfl_xor(lsum, off, 32);
  if ((tid & 31) == 0) red[tid >> 5] = lsum;
  __syncthreads();
  float bs = 0.f;
  #pragma unroll
  for (int w = 0; w < 8; ++w) bs += red[w];
  const float invs = 1.f / bs;

  float* arow = attn + ((long)b * Q_ + q) * K_;
  #pragma unroll
  for (int j = 0; j < 4; ++j)
    arow[tid + j * 256] = p[j] * invs;   // fully coalesced per j
}

// ---------------------------------------------------------------------------
// inputs: queries[B,Q,DQ], keys[B,K,DK], values[B,K,DV], valid_lens[B],
//         Wq[DQ,H], Wk[DK,H], wv[H]     -> out[B,Q,DV]
// ---------------------------------------------------------------------------
extern "C" void kernel_launch(void* const* d_in, const int* in_sizes, int n_in,
                              void* d_out, int out_size, void* d_ws, size_t ws_size,
                              hipStream_t stream) {
  (void)in_sizes; (void)n_in; (void)out_size; (void)ws_size;
  const float* queries = (const float*)d_in[0];
  const float* keys    = (const float*)d_in[1];
  const float* values  = (const float*)d_in[2];
  const int*   vlen    = (const int*)d_in[3];
  const float* Wq      = (const float*)d_in[4];
  const float* Wk      = (const float*)d_in[5];
  const float* wv      = (const float*)d_in[6];
  float* out = (float*)d_out;

  float* qf   = (float*)d_ws;                 // [B*Q, H]   = 512KB
  float* kf   = qf + (long)B_ * Q_ * H_;      // [B*K, H]   = 4MB
  float* attn = kf + (long)B_ * K_ * H_;      // [B, Q, K]  = 2MB

  const dim3 wave(32);

  // qf = queries @ Wq  (M = B*Q = 512 rows, KD = DQ, N = H; 32-wide N tiles)
  gemm_wmma_f32<H_, DQ_><<<dim3((B_ * Q_) / 16, H_ / 32, 1), wave, 0, stream>>>(
      queries, Wq, qf, 0, 0, 0);

  // kf = keys @ Wk  (M = B*K = 4096 rows, KD = DK, N = H)
  gemm_wmma_f32<H_, DK_><<<dim3((B_ * K_) / 16, H_ / 32, 1), wave, 0, stream>>>(
      keys, Wk, kf, 0, 0, 0);

  // attn = softmax(mask(sum_h wv*tanh(qf+kf)))
  scores_softmax_kernel<<<dim3(Q_, B_), 256, 0, stream>>>(qf, kf, wv, vlen, attn);

  // out[b] = attn[b] @ values[b]   (batched: M=Q, KD=K, N=DV)
  gemm_wmma_f32<DV_, K_><<<dim3(Q_ / 16, DV_ / 32, B_), wave, 0, stream>>>(
      attn, values, out, (long)Q_ * K_, (long)K_ * DV_, (long)Q_ * DV_);
}